// NonstationaryGaussianSpectralMixtureKernel_14534169329757
// MI455X (gfx1250) — compile-verified
//
#include <hip/hip_runtime.h>
#include <hip/hip_bf16.h>
#include <math.h>

// Nonstationary Gaussian spectral-mixture kernel, MI455X (gfx1250, wave32).
//
// Phase 1 (negligible cost): per-point MLP features, folded into
//   u[q] = w*sqrt(s)*2^0.25*cos(2*pi*phase),  v[q] = ...*sin(...),  s2[q] = s^2,
//   plus ||p||^2.  This removes sqrt/cos/sin/w from the O(N*M*Q) loop.
//
// Phase 2 (dominant, transcendental-bound): per 16x16 (i,j) tile,
//   d2 = ||x||^2 + ||y||^2 - 2 x.y  via V_WMMA_F32_16X16X4_F32
//   (A = [-2x0,-2x1,-2x2,1], B = [y0,y1,y2,||y||^2], C = ||x||^2),
//   then per q: K += (uX*uY + vX*vY) * inv * exp2(-d2*log2e*inv),
//   inv = v_rcp_f32(sX^2 + sY^2), exp via v_exp_f32.

#define QMIX 8
#define LATENT 64

typedef __attribute__((ext_vector_type(2))) float v2f;
typedef __attribute__((ext_vector_type(8))) float v8f;

__device__ __forceinline__ float softplus_f(float z) {
    // stable: max(z,0) + log1p(exp(-|z|))
    return fmaxf(z, 0.0f) + log1pf(__expf(-fabsf(z)));
}

__device__ __forceinline__ float selu_f(float z) {
    const float lam = 1.0507009873554805f;
    const float alp = 1.6732632423543772f;
    return z > 0.0f ? lam * z : lam * alp * (__expf(z) - 1.0f);
}

__global__ void __launch_bounds__(128)
point_feat_kernel(const float* __restrict__ P, int n,
                  const float* __restrict__ W1, const float* __restrict__ b1,
                  const float* __restrict__ Ww, const float* __restrict__ bw,
                  const float* __restrict__ Wf, const float* __restrict__ bf,
                  const float* __restrict__ Ws, const float* __restrict__ bs,
                  float* __restrict__ feat,   // [n][24] = u[8], v[8], s2[8]
                  float* __restrict__ norms)  // [n] = ||p||^2
{
    int i = blockIdx.x * blockDim.x + threadIdx.x;
    if (i >= n) return;
    float p0 = P[(size_t)i * 3 + 0];
    float p1 = P[(size_t)i * 3 + 1];
    float p2 = P[(size_t)i * 3 + 2];
    norms[i] = p0 * p0 + p1 * p1 + p2 * p2;

    float h[LATENT];
#pragma unroll
    for (int l = 0; l < LATENT; ++l) {
        float z = b1[l] + W1[l * 3 + 0] * p0 + W1[l * 3 + 1] * p1 + W1[l * 3 + 2] * p2;
        h[l] = selu_f(z);
    }

    for (int q = 0; q < QMIX; ++q) {
        float zw = bw[q];
        float zs = bs[q];
        float zf0 = bf[q * 3 + 0];
        float zf1 = bf[q * 3 + 1];
        float zf2 = bf[q * 3 + 2];
#pragma unroll
        for (int l = 0; l < LATENT; ++l) {
            float hl = h[l];
            zw  = fmaf(Ww[q * LATENT + l], hl, zw);
            zs  = fmaf(Ws[q * LATENT + l], hl, zs);
            zf0 = fmaf(Wf[(q * 3 + 0) * LATENT + l], hl, zf0);
            zf1 = fmaf(Wf[(q * 3 + 1) * LATENT + l], hl, zf1);
            zf2 = fmaf(Wf[(q * 3 + 2) * LATENT + l], hl, zf2);
        }
        float w = softplus_f(zw);
        float s = softplus_f(zs);
        float f0 = softplus_f(zf0), f1 = softplus_f(zf1), f2 = softplus_f(zf2);
        float ph  = f0 * p0 + f1 * p1 + f2 * p2;
        float ang = 6.283185307179586f * ph;
        // amp = w * sqrt(s) * 2^(1/4); product of two amps carries the sqrt(2).
        float amp = w * __builtin_amdgcn_sqrtf(s) * 1.189207115002721f;
        feat[(size_t)i * 24 + q]      = amp * __cosf(ang);  // u
        feat[(size_t)i * 24 + 8 + q]  = amp * __sinf(ang);  // v
        feat[(size_t)i * 24 + 16 + q] = s * s;              // s^2
    }
}

__device__ __forceinline__ float termf(float ux, float vx, float s2x,
                                       float uy, float vy, float s2y,
                                       float d2l) {
    float s2  = s2x + s2y;
    float inv = __builtin_amdgcn_rcpf(s2);                  // v_rcp_f32
    float e   = __builtin_amdgcn_exp2f(-d2l * inv);         // v_exp_f32
    float cr  = fmaf(ux, uy, vx * vy);                      // uXuY + vXvY
    return cr * inv * e;
}

__global__ void __launch_bounds__(256)
pair_kernel(const float* __restrict__ X, const float* __restrict__ Y,
            const float* __restrict__ xfeat, const float* __restrict__ yfeat,
            const float* __restrict__ xn, const float* __restrict__ yn,
            float* __restrict__ out, int M, int tilesJ, int tiles)
{
    int wave = threadIdx.x >> 5;
    int lane = threadIdx.x & 31;
    int tile = blockIdx.x * 8 + wave;
    if (tile >= tiles) return;               // uniform per wave: EXEC stays all-1s
    int ti = tile / tilesJ;
    int tj = tile - ti * tilesJ;
    int i0 = ti << 4;
    int j0 = tj << 4;
    int lg = lane >> 4;                      // lane group (0: lanes 0-15, 1: 16-31)
    int hf = lane & 15;

    // A matrix 16x4 (f32): VGPR0 = K{0|2}, VGPR1 = K{1|3}; row M = lane%16.
    const float* xr = X + (size_t)(i0 + hf) * 3;
    float ax0 = -2.0f * xr[0], ax1 = -2.0f * xr[1], ax2 = -2.0f * xr[2];
    v2f a;
    a.x = lg ? ax2 : ax0;
    a.y = lg ? 1.0f : ax1;

    // B matrix 4x16 (f32): VGPR0 = K{0|2}, VGPR1 = K{1|3}; col N = lane%16.
    const float* yr = Y + (size_t)(j0 + hf) * 3;
    float by0 = yr[0], by1 = yr[1], by2 = yr[2];
    float yn2 = yn[j0 + hf];
    v2f b;
    b.x = lg ? by2 : by0;
    b.y = lg ? yn2 : by1;

    // C matrix: c[r] = ||x_{i0+r+8*lg}||^2  (row-broadcast term of d2)
    const float4* cn4 = (const float4*)(xn + i0 + 8 * lg);
    float4 c0 = cn4[0], c1 = cn4[1];
    v8f c;
    c[0] = c0.x; c[1] = c0.y; c[2] = c0.z; c[3] = c0.w;
    c[4] = c1.x; c[5] = c1.y; c[6] = c1.z; c[7] = c1.w;

    // D[r] @ (row = i0 + r + 8*lg, col = j0 + lane%16) = squared distance
    v8f d = __builtin_amdgcn_wmma_f32_16x16x4_f32(
        false, a, false, b, (short)0, c, false, false);

    // This lane's column features (resident for the whole tile).
    const float4* yf4 = (const float4*)(yfeat + (size_t)(j0 + hf) * 24);
    float4 yu0 = yf4[0], yu1 = yf4[1];
    float4 yv0 = yf4[2], yv1 = yf4[3];
    float4 ys0 = yf4[4], ys1 = yf4[5];

    const float LOG2E = 1.4426950408889634f;
#pragma unroll
    for (int r = 0; r < 8; ++r) {
        int m = i0 + 8 * lg + r;
        const float4* xf4 = (const float4*)(xfeat + (size_t)m * 24);
        float4 xu0 = xf4[0], xu1 = xf4[1];
        float4 xv0 = xf4[2], xv1 = xf4[3];
        float4 xs0 = xf4[4], xs1 = xf4[5];
        float d2l = d[r] * LOG2E;
        float acc;
        acc  = termf(xu0.x, xv0.x, xs0.x, yu0.x, yv0.x, ys0.x, d2l);
        acc += termf(xu0.y, xv0.y, xs0.y, yu0.y, yv0.y, ys0.y, d2l);
        acc += termf(xu0.z, xv0.z, xs0.z, yu0.z, yv0.z, ys0.z, d2l);
        acc += termf(xu0.w, xv0.w, xs0.w, yu0.w, yv0.w, ys0.w, d2l);
        acc += termf(xu1.x, xv1.x, xs1.x, yu1.x, yv1.x, ys1.x, d2l);
        acc += termf(xu1.y, xv1.y, xs1.y, yu1.y, yv1.y, ys1.y, d2l);
        acc += termf(xu1.z, xv1.z, xs1.z, yu1.z, yv1.z, ys1.z, d2l);
        acc += termf(xu1.w, xv1.w, xs1.w, yu1.w, yv1.w, ys1.w, d2l);
        out[(size_t)m * M + j0 + hf] = acc;
    }
}

extern "C" void kernel_launch(void* const* d_in, const int* in_sizes, int n_in,
                              void* d_out, int out_size, void* d_ws, size_t ws_size,
                              hipStream_t stream) {
    const float* x  = (const float*)d_in[0];
    const float* y  = (const float*)d_in[1];
    const float* W1 = (const float*)d_in[2];
    const float* b1 = (const float*)d_in[3];
    const float* Ww = (const float*)d_in[4];
    const float* bw = (const float*)d_in[5];
    const float* Wf = (const float*)d_in[6];
    const float* bf = (const float*)d_in[7];
    const float* Ws = (const float*)d_in[8];
    const float* bs = (const float*)d_in[9];

    int N = in_sizes[0] / 3;
    int M = in_sizes[1] / 3;

    float* xfeat = (float*)d_ws;                 // N*24
    float* yfeat = xfeat + (size_t)N * 24;       // M*24
    float* xn    = yfeat + (size_t)M * 24;       // N
    float* yn    = xn + N;                       // M

    point_feat_kernel<<<(N + 127) / 128, 128, 0, stream>>>(
        x, N, W1, b1, Ww, bw, Wf, bf, Ws, bs, xfeat, xn);
    point_feat_kernel<<<(M + 127) / 128, 128, 0, stream>>>(
        y, M, W1, b1, Ww, bw, Wf, bf, Ws, bs, yfeat, yn);

    int tilesI = N >> 4;
    int tilesJ = M >> 4;
    int tiles  = tilesI * tilesJ;
    pair_kernel<<<(tiles + 7) / 8, 256, 0, stream>>>(
        x, y, xfeat, yfeat, xn, yn, (float*)d_out, M, tilesJ, tiles);
}